// GCNReg_11424613007554
// MI455X (gfx1250) — compile-verified
//
#include <hip/hip_runtime.h>

typedef __attribute__((ext_vector_type(2))) float v2f;
typedef __attribute__((ext_vector_type(8))) float v8f;

namespace {
constexpr int kNodes  = 100000;
constexpr int kEdges  = 3200000;
constexpr int kFeat   = 128;
constexpr int kH1     = 32;
constexpr int kH2     = 16;
constexpr int kGraphs = 64;
constexpr int kMTiles = kNodes / 16;  // 6250, exact
}

// ---------------- utility fills ----------------
__global__ void fill_f32(float* __restrict__ p, float v, int n) {
  int i = blockIdx.x * blockDim.x + threadIdx.x;
  if (i < n) p[i] = v;
}

// ---------------- degree + rsqrt ----------------
__global__ void degree_k(const long long* __restrict__ dst, float* __restrict__ deg, int e) {
  int i = blockIdx.x * blockDim.x + threadIdx.x;
  if (i < e) atomicAdd(&deg[(int)dst[i]], 1.0f);
}

__global__ void rsqrt_k(const float* __restrict__ deg, float* __restrict__ dis, int n) {
  int i = blockIdx.x * blockDim.x + threadIdx.x;
  if (i < n) dis[i] = rsqrtf(deg[i]);  // deg >= 1 (self-loops), matches reference
}

// ---------------- GEMM 1: xw = x @ W1, f32 WMMA 16x16x4 ----------------
// One wave per 16-row M tile; N=32 -> two 16-wide accumulators; K=128 in steps of 4.
__global__ __launch_bounds__(256) void gemm_xw1(const float* __restrict__ x,
                                                const float* __restrict__ W,
                                                float* __restrict__ xw) {
  constexpr int LDW = kFeat + 2;  // 130: padded transpose, conflict-free ds_load_b64
  __shared__ float WsT[kH1 * LDW];
  for (int idx = threadIdx.x; idx < kFeat * kH1; idx += 256) {
    int k = idx / kH1, n = idx % kH1;      // W is [128][32] row-major
    WsT[n * LDW + k] = W[idx];
  }
  __syncthreads();

  const int wave = threadIdx.x >> 5;
  const int lane = threadIdx.x & 31;
  int tile = blockIdx.x * 8 + wave;
  if (tile >= kMTiles) tile = kMTiles - 1;  // idempotent tail duplication keeps EXEC all-ones

  const int m   = lane & 15;
  const int kkb = (lane >> 4) << 1;  // lanes 0-15: K sub 0/1, lanes 16-31: K sub 2/3 (ISA A layout)
  const float* xrow = x + (size_t)(tile * 16 + m) * kFeat;
  const float* bp0  = &WsT[m * LDW];
  const float* bp1  = &WsT[(m + 16) * LDW];

  v8f acc0 = {0.f, 0.f, 0.f, 0.f, 0.f, 0.f, 0.f, 0.f};
  v8f acc1 = acc0;
  for (int k4 = 0; k4 < kFeat; k4 += 4) {
    v2f a, b0, b1;
    a.x  = xrow[k4 + kkb];  a.y  = xrow[k4 + kkb + 1];
    b0.x = bp0[k4 + kkb];   b0.y = bp0[k4 + kkb + 1];
    b1.x = bp1[k4 + kkb];   b1.y = bp1[k4 + kkb + 1];
    acc0 = __builtin_amdgcn_wmma_f32_16x16x4_f32(false, a, false, b0, (short)0, acc0, false, false);
    acc1 = __builtin_amdgcn_wmma_f32_16x16x4_f32(false, a, false, b1, (short)0, acc1, false, false);
  }
  const int rbase = (lane >> 4) << 3;  // C layout: lanes 16-31 hold rows M = r + 8
  for (int r = 0; r < 8; ++r) {
    size_t row = (size_t)(tile * 16 + rbase + r);
    xw[row * kH1 + m]      = acc0[r];
    xw[row * kH1 + m + 16] = acc1[r];
  }
}

// ---------------- GEMM 2: hw = h @ W2 (K=32, N=16) ----------------
__global__ __launch_bounds__(256) void gemm_hw2(const float* __restrict__ h,
                                                const float* __restrict__ W,
                                                float* __restrict__ hw) {
  constexpr int LDW = kH1 + 2;  // 34
  __shared__ float WsT[kH2 * LDW];
  for (int idx = threadIdx.x; idx < kH1 * kH2; idx += 256) {
    int k = idx / kH2, n = idx % kH2;      // W is [32][16] row-major
    WsT[n * LDW + k] = W[idx];
  }
  __syncthreads();

  const int wave = threadIdx.x >> 5;
  const int lane = threadIdx.x & 31;
  int tile = blockIdx.x * 8 + wave;
  if (tile >= kMTiles) tile = kMTiles - 1;

  const int m   = lane & 15;
  const int kkb = (lane >> 4) << 1;
  const float* hrow = h + (size_t)(tile * 16 + m) * kH1;
  const float* bp   = &WsT[m * LDW];

  v8f acc = {0.f, 0.f, 0.f, 0.f, 0.f, 0.f, 0.f, 0.f};
  for (int k4 = 0; k4 < kH1; k4 += 4) {
    v2f a, b;
    a.x = hrow[k4 + kkb];  a.y = hrow[k4 + kkb + 1];
    b.x = bp[k4 + kkb];    b.y = bp[k4 + kkb + 1];
    acc = __builtin_amdgcn_wmma_f32_16x16x4_f32(false, a, false, b, (short)0, acc, false, false);
  }
  const int rbase = (lane >> 4) << 3;
  for (int r = 0; r < 8; ++r) {
    size_t row = (size_t)(tile * 16 + rbase + r);
    hw[row * kH2 + m] = acc[r];
  }
}

// ---------------- edge scatter: agg[dst] += msg[src] * dis[src]*dis[dst] ----------------
template <int LOG2F>
__global__ void scatter_k(const long long* __restrict__ src, const long long* __restrict__ dst,
                          const float* __restrict__ dis, const float* __restrict__ msg,
                          float* __restrict__ agg, int nwork) {
  int t = blockIdx.x * blockDim.x + threadIdx.x;
  if (t >= nwork) return;
  int e = t >> LOG2F;
  int f = t & ((1 << LOG2F) - 1);
  int s = (int)src[e];
  int d = (int)dst[e];
  float w = dis[s] * dis[d];
  atomicAdd(&agg[((size_t)d << LOG2F) + f], msg[((size_t)s << LOG2F) + f] * w);
}

// ---------------- epilogue 1: fold self-loop + bias + ReLU (in place on agg1) ----------------
__global__ void epi1_k(const float* __restrict__ xw, const float* __restrict__ dis,
                       const float* __restrict__ b1, float* __restrict__ agg1, int nwork) {
  int t = blockIdx.x * blockDim.x + threadIdx.x;
  if (t >= nwork) return;
  int i = t >> 5;
  int f = t & 31;
  float d2 = dis[i] * dis[i];
  float v  = agg1[t] + xw[t] * d2 + b1[f];
  agg1[t]  = fmaxf(v, 0.0f);
}

// ---------------- epilogue 2 + mean-pool accumulation ----------------
__global__ void epi2_pool_k(const float* __restrict__ hw, const float* __restrict__ dis,
                            const float* __restrict__ b2, const long long* __restrict__ batch,
                            const float* __restrict__ agg2, float* __restrict__ gsum,
                            float* __restrict__ gcnt, int nwork) {
  int t = blockIdx.x * blockDim.x + threadIdx.x;
  if (t >= nwork) return;
  int i = t >> 4;
  int f = t & 15;
  float d2 = dis[i] * dis[i];
  float v  = agg2[t] + hw[t] * d2 + b2[f];
  int g = (int)batch[i];
  atomicAdd(&gsum[g * kH2 + f], v);
  if (f == 0) atomicAdd(&gcnt[g], 1.0f);
}

// ---------------- head: out[g] = mean(h2) @ lin_w + lin_b ----------------
__global__ void head_k(const float* __restrict__ gsum, const float* __restrict__ gcnt,
                       const float* __restrict__ lw, const float* __restrict__ lb,
                       float* __restrict__ out) {
  int g = threadIdx.x;
  if (g >= kGraphs) return;
  float inv = 1.0f / fmaxf(gcnt[g], 1.0f);
  float acc = 0.0f;
#pragma unroll
  for (int f = 0; f < kH2; ++f) acc += gsum[g * kH2 + f] * lw[f];
  out[g] = acc * inv + lb[0];
}

extern "C" void kernel_launch(void* const* d_in, const int* in_sizes, int n_in,
                              void* d_out, int out_size, void* d_ws, size_t ws_size,
                              hipStream_t stream) {
  const float*     x     = (const float*)d_in[0];
  const long long* ei    = (const long long*)d_in[1];  // int64 [2, E]
  const long long* batch = (const long long*)d_in[2];  // int64 [N]
  const float*     W1    = (const float*)d_in[3];
  const float*     b1    = (const float*)d_in[4];
  const float*     W2    = (const float*)d_in[5];
  const float*     b2    = (const float*)d_in[6];
  const float*     lw    = (const float*)d_in[7];
  const float*     lb    = (const float*)d_in[8];
  float*           out   = (float*)d_out;

  const long long* src = ei;
  const long long* dst = ei + kEdges;

  // workspace layout (floats); total ~9.8M floats ~= 39.2 MB
  float* ws   = (float*)d_ws;
  float* xw   = ws;                               // N*32
  float* hw   = xw   + (size_t)kNodes * kH1;      // N*16
  float* agg1 = hw   + (size_t)kNodes * kH2;      // N*32  (zeroed)
  float* agg2 = agg1 + (size_t)kNodes * kH1;      // N*16  (zeroed)
  float* gsum = agg2 + (size_t)kNodes * kH2;      // 64*16 (zeroed)
  float* gcnt = gsum + kGraphs * kH2;             // 64    (zeroed)
  float* deg  = gcnt + kGraphs;                   // N (init 1.0 = self-loop)
  float* dis  = deg  + kNodes;                    // N

  const int zn = kNodes * kH1 + kNodes * kH2 + kGraphs * kH2 + kGraphs;  // agg1..gcnt contiguous

  fill_f32<<<(zn + 255) / 256, 256, 0, stream>>>(agg1, 0.0f, zn);
  fill_f32<<<(kNodes + 255) / 256, 256, 0, stream>>>(deg, 1.0f, kNodes);

  degree_k<<<(kEdges + 255) / 256, 256, 0, stream>>>(dst, deg, kEdges);
  rsqrt_k<<<(kNodes + 255) / 256, 256, 0, stream>>>(deg, dis, kNodes);

  // conv1
  gemm_xw1<<<(kMTiles + 7) / 8, 256, 0, stream>>>(x, W1, xw);
  {
    const int nwork = kEdges * kH1;  // 102.4M lanes, one (edge,feat) each
    scatter_k<5><<<(nwork + 255) / 256, 256, 0, stream>>>(src, dst, dis, xw, agg1, nwork);
  }
  epi1_k<<<(kNodes * kH1 + 255) / 256, 256, 0, stream>>>(xw, dis, b1, agg1, kNodes * kH1);

  // conv2 (input h lives in agg1)
  gemm_hw2<<<(kMTiles + 7) / 8, 256, 0, stream>>>(agg1, W2, hw);
  {
    const int nwork = kEdges * kH2;  // 51.2M lanes
    scatter_k<4><<<(nwork + 255) / 256, 256, 0, stream>>>(src, dst, dis, hw, agg2, nwork);
  }
  epi2_pool_k<<<(kNodes * kH2 + 255) / 256, 256, 0, stream>>>(hw, dis, b2, batch, agg2,
                                                              gsum, gcnt, kNodes * kH2);

  head_k<<<1, 64, 0, stream>>>(gsum, gcnt, lw, lb, out);
}